// WeightedNeighborSampler_17583596110457
// MI455X (gfx1250) — compile-verified
//
#include <hip/hip_runtime.h>
#include <hip/hip_bf16.h>
#include <stdint.h>

// WeightedNeighborSampler for MI455X (gfx1250, wave32).
//
// Inputs (setup_inputs order):
//   d_in[0] adj_weight : float32 [N_NODES * 128]
//   d_in[1] adj_info   : int32   [N_NODES * 128]
//   d_in[2] ids        : int32   [BATCH]
//   d_in[3] num_samples: int32   [1]   (== 25)
// Output: int32 [BATCH * num_samples] sampled neighbor ids.
//
// One wave32 per batch row. The neighbor-id row (512B) is staged into LDS with
// per-lane global_load_async_to_lds_b128 (CDNA5 async path, ASYNCcnt),
// overlapped with the weight-row load + CDF scan; the final sample gather then
// hits LDS instead of a second global gather.

#define MAX_DEG 128
#define WAVES_PER_BLOCK 8

// Deterministic per-(row,sample) uniform in (0,1): splitmix64-style mix.
__device__ __forceinline__ float rng_u01(unsigned row, unsigned s) {
    unsigned long long x =
        (((unsigned long long)row) << 32) ^ (0x9E3779B97F4A7C15ULL * (unsigned long long)(s + 1u));
    x ^= x >> 33; x *= 0xFF51AFD7ED558CCDULL;
    x ^= x >> 33; x *= 0xC4CEB9FE1A85EC53ULL;
    x ^= x >> 33;
    // top 24 bits -> (0,1)
    return ((float)((unsigned)(x >> 40)) + 0.5f) * (1.0f / 16777216.0f);
}

__global__ void __launch_bounds__(32 * WAVES_PER_BLOCK)
wns_sample_kernel(const float* __restrict__ adj_weight,
                  const int*   __restrict__ adj_info,
                  const int*   __restrict__ ids,
                  const int*   __restrict__ ns_ptr,
                  int*         __restrict__ out,
                  int batch) {
    __shared__ __align__(16) float s_cdf[WAVES_PER_BLOCK][MAX_DEG]; // inclusive CDF
    __shared__ __align__(16) int   s_nbr[WAVES_PER_BLOCK][MAX_DEG]; // neighbor ids

    const int lane = (int)(threadIdx.x & 31u);
    const int wv   = (int)(threadIdx.x >> 5u);

    int  b     = (int)blockIdx.x * WAVES_PER_BLOCK + wv;
    bool valid = (b < batch);
    int  bc    = valid ? b : (batch - 1);   // clamp: keep all waves convergent

    const long long  node = (long long)ids[bc];
    const float* wrow = adj_weight + (size_t)node * MAX_DEG;
    const int*   irow = adj_info   + (size_t)node * MAX_DEG;

    // ---- CDNA5 async path: stage the 512B neighbor-id row into LDS.
    // 32 lanes x b128 = 512B. Tracked by ASYNCcnt; overlaps the CDF build.
    {
        unsigned           lds_dst = (unsigned)(size_t)(&s_nbr[wv][lane * 4]);
        unsigned long long gsrc    = (unsigned long long)(irow + lane * 4);
        asm volatile("global_load_async_to_lds_b128 %0, %1, off"
                     :: "v"(lds_dst), "v"(gsrc)
                     : "memory");
    }

    // ---- Weight row: coalesced float4 per lane (512B/row), lane-local scan.
    float4 w = *(const float4*)(wrow + lane * 4);
    float c0 = w.x;
    float c1 = c0 + w.y;
    float c2 = c1 + w.z;
    float c3 = c2 + w.w;

    // Wave32 inclusive scan of per-lane sums (Hillis-Steele over 32 lanes).
    float incl = c3;
#pragma unroll
    for (int d = 1; d < 32; d <<= 1) {
        float t = __shfl_up(incl, d, 32);
        if (lane >= d) incl += t;
    }
    float excl  = incl - c3;
    float total = __shfl(incl, 31, 32);

    s_cdf[wv][lane * 4 + 0] = excl + c0;
    s_cdf[wv][lane * 4 + 1] = excl + c1;
    s_cdf[wv][lane * 4 + 2] = excl + c2;
    s_cdf[wv][lane * 4 + 3] = excl + c3;

    // Wait for the async id-row DMA; barrier keeps the compiler from hoisting
    // the LDS gather above the wait. (Intra-wave LDS store->load is in-order.)
    asm volatile("s_wait_asynccnt 0" ::: "memory");
    __builtin_amdgcn_wave_barrier();

    const int ns = ns_ptr[0];

    // Each of the first ns lanes draws one sample (ns=25 < 32; loop for safety).
    for (int j = lane; j < ns; j += 32) {
        float target = rng_u01((unsigned)bc, (unsigned)j) * total;
        // First index with cdf[idx] >= target (7-step binary search in LDS).
        int lo = 0, hi = MAX_DEG - 1;
#pragma unroll
        for (int it = 0; it < 7; ++it) {
            int mid = (lo + hi) >> 1;
            if (s_cdf[wv][mid] < target) lo = mid + 1; else hi = mid;
        }
        if (valid) out[(size_t)b * ns + j] = s_nbr[wv][lo];
    }
}

extern "C" void kernel_launch(void* const* d_in, const int* in_sizes, int n_in,
                              void* d_out, int out_size, void* d_ws, size_t ws_size,
                              hipStream_t stream) {
    (void)n_in; (void)out_size; (void)d_ws; (void)ws_size;
    const float* adj_weight = (const float*)d_in[0];
    const int*   adj_info   = (const int*)d_in[1];
    const int*   ids        = (const int*)d_in[2];
    const int*   ns_ptr     = (const int*)d_in[3];
    int*         out        = (int*)d_out;

    const int batch  = in_sizes[2];
    const int blocks = (batch + WAVES_PER_BLOCK - 1) / WAVES_PER_BLOCK;

    wns_sample_kernel<<<blocks, 32 * WAVES_PER_BLOCK, 0, stream>>>(
        adj_weight, adj_info, ids, ns_ptr, out, batch);
}